// GNNBlock_75720273428864
// MI455X (gfx1250) — compile-verified
//
#include <hip/hip_runtime.h>
#include <hip/hip_bf16.h>

#define N_NODES 50000
#define N_EDGES 600000
#define D 128
#define P_DROP 0.1f
#define BN_EPS 1e-5f

typedef __attribute__((ext_vector_type(2))) float v2f;
typedef __attribute__((ext_vector_type(8))) float v8f;

static __device__ __forceinline__ void atomic_add_f32(float* p, float v) {
    __hip_atomic_fetch_add(p, v, __ATOMIC_RELAXED, __HIP_MEMORY_SCOPE_AGENT);
}

// ---------------------------------------------------------------------------
// K0: deg[i] = 1.0 (self loop), zero the per-channel sum/sumsq accumulators
// ---------------------------------------------------------------------------
__global__ void k_init(float* __restrict__ deg, float* __restrict__ sums) {
    int i = blockIdx.x * blockDim.x + threadIdx.x;
    if (i < N_NODES) deg[i] = 1.0f;
    if (i < 2 * D)   sums[i] = 0.0f;
}

// ---------------------------------------------------------------------------
// K1: deg[dst] += 1 per edge (coalesced b64 index loads, f32 HW atomics)
// ---------------------------------------------------------------------------
__global__ void k_degree(const long long* __restrict__ ei, float* __restrict__ deg) {
    int e = blockIdx.x * blockDim.x + threadIdx.x;
    if (e >= N_EDGES) return;
    long long dst = ei[(long long)N_EDGES + e];
    atomic_add_f32(&deg[dst], 1.0f);
}

// ---------------------------------------------------------------------------
// K2: dinv = rsqrt(deg)   (in place; deg >= 1 always)
// ---------------------------------------------------------------------------
__global__ void k_dinv(float* __restrict__ deg) {
    int i = blockIdx.x * blockDim.x + threadIdx.x;
    if (i < N_NODES) deg[i] = rsqrtf(deg[i]);
}

// ---------------------------------------------------------------------------
// K3: WMMA f32 GEMM: y = (x @ W) * dinv[row]; acc = y (self-loop init).
// 256-thread block = 8 waves; each wave owns a 16x128 output strip.
// Weight staged TRANSPOSED in LDS with XOR swizzle:
//   wt[n*128 + (k ^ ((n&15)<<2))]
// -> B fragment = one conflict-free ds_load_b64 directly into an aligned
//    VGPR pair (no repack moves before v_wmma).  Exactly 64 KB LDS.
// ---------------------------------------------------------------------------
__global__ void __launch_bounds__(256)
k_gemm(const float* __restrict__ x, const float* __restrict__ w,
       const float* __restrict__ dinv, float* __restrict__ y,
       float* __restrict__ acc) {
    __shared__ float wt[D * D];                 // 64 KB

    // stage: read w row-major (float4, coalesced), scatter transposed+swizzled
    for (int f4 = threadIdx.x; f4 < (D * D) / 4; f4 += 256) {
        const int f = f4 * 4;
        const int k = f >> 7;                   // row of w
        const int n0 = f & 127;                 // 4 consecutive columns
        const float4 v = *(const float4*)(w + f);
        wt[((n0 + 0) << 7) + (k ^ (((n0 + 0) & 15) << 2))] = v.x;
        wt[((n0 + 1) << 7) + (k ^ (((n0 + 1) & 15) << 2))] = v.y;
        wt[((n0 + 2) << 7) + (k ^ (((n0 + 2) & 15) << 2))] = v.z;
        wt[((n0 + 3) << 7) + (k ^ (((n0 + 3) & 15) << 2))] = v.w;
    }
    __syncthreads();

    const int wave = threadIdx.x >> 5;
    const int lane = threadIdx.x & 31;
    const int half = lane >> 4;                 // 0: K pair {0,1}, 1: {2,3}
    const int lm   = lane & 15;
    const int m0   = (blockIdx.x * 8 + wave) * 16;
    if (m0 >= N_NODES) return;                  // tail waves idle (after barrier)

    v8f c[8] = {};
    const float* xrow = x + (size_t)(m0 + lm) * D;
    const int swz = lm << 2;                    // (n&15)<<2, same for every nt

    for (int k0 = 0; k0 < D; k0 += 4) {
        const int ko = k0 + 2 * half;           // even
        // A frag (16x4 f32): lanes 0-15 -> K={k0,k0+1}, lanes 16-31 -> K={k0+2,k0+3}
        const v2f a = *(const v2f*)(xrow + ko);
        const int kswz = ko ^ swz;              // even -> 8B-aligned b64 loads
#pragma unroll
        for (int nt = 0; nt < 8; ++nt) {
            const v2f b = *(const v2f*)(wt + ((nt * 16 + lm) << 7) + kswz);
            c[nt] = __builtin_amdgcn_wmma_f32_16x16x4_f32(
                false, a, false, b, (short)0, c[nt], false, false);
        }
    }

    // epilogue: scale rows by dinv, store y and acc (self-loop term)
    float dv[8];
#pragma unroll
    for (int r = 0; r < 8; ++r) dv[r] = dinv[m0 + r + 8 * half];
#pragma unroll
    for (int nt = 0; nt < 8; ++nt) {
#pragma unroll
        for (int r = 0; r < 8; ++r) {
            const int m = m0 + r + 8 * half;
            const int n = nt * 16 + lm;
            const float v = c[nt][r] * dv[r];
            const size_t idx = (size_t)m * D + n;
            y[idx]   = v;
            acc[idx] = v;
        }
    }
}

// ---------------------------------------------------------------------------
// K4: edge scatter-add: acc[dst] += y[src].  One wave per edge, float4/lane.
// Edge id is wave-uniform -> readfirstlane so src/dst come from scalar loads.
// y and acc both resident in the 192 MB L2 -> L2-rate gather + atomics.
// ---------------------------------------------------------------------------
__global__ void __launch_bounds__(256)
k_edge_agg(const long long* __restrict__ ei, const float* __restrict__ y,
           float* __restrict__ acc) {
    const int e = __builtin_amdgcn_readfirstlane(
        blockIdx.x * (blockDim.x >> 5) + (threadIdx.x >> 5));
    if (e >= N_EDGES) return;
    const int lane = threadIdx.x & 31;
    const long long src = ei[e];
    const long long dst = ei[(long long)N_EDGES + e];
    const float4 v = *(const float4*)(y + (size_t)src * D + lane * 4);
    float* d = acc + (size_t)dst * D + lane * 4;
    atomic_add_f32(d + 0, v.x);
    atomic_add_f32(d + 1, v.y);
    atomic_add_f32(d + 2, v.z);
    atomic_add_f32(d + 3, v.w);
}

// ---------------------------------------------------------------------------
// K5: z = relu(dinv[r]*acc[r] + bias); write z; per-channel sum & sumsq.
// 128 threads = 4 waves; lane owns 4 channels (float4), waves stride rows.
// Cross-wave LDS reduction -> 2 global atomics per channel per block.
// ---------------------------------------------------------------------------
#define BN_ROWS 125
__global__ void __launch_bounds__(128)
k_bn_stats(const float* __restrict__ acc, const float* __restrict__ dinv,
           const float* __restrict__ bias, float* __restrict__ z,
           float* __restrict__ sums) {
    __shared__ float red[4 * 2 * D];            // [rowgroup][{sum,sumsq}][channel]
    const int lane = threadIdx.x & 31;
    const int rg   = threadIdx.x >> 5;          // 0..3
    const int c0   = lane * 4;
    const int r0 = blockIdx.x * BN_ROWS;
    const int r1 = min(r0 + BN_ROWS, N_NODES);
    const float4 b = *(const float4*)(bias + c0);
    float4 s  = {0.f, 0.f, 0.f, 0.f};
    float4 s2 = {0.f, 0.f, 0.f, 0.f};
    for (int r = r0 + rg; r < r1; r += 4) {
        const float di = dinv[r];
        float4 v = *(const float4*)(acc + (size_t)r * D + c0);
        v.x = fmaxf(fmaf(v.x, di, b.x), 0.f);
        v.y = fmaxf(fmaf(v.y, di, b.y), 0.f);
        v.z = fmaxf(fmaf(v.z, di, b.z), 0.f);
        v.w = fmaxf(fmaf(v.w, di, b.w), 0.f);
        *(float4*)(z + (size_t)r * D + c0) = v;
        s.x += v.x; s.y += v.y; s.z += v.z; s.w += v.w;
        s2.x += v.x * v.x; s2.y += v.y * v.y; s2.z += v.z * v.z; s2.w += v.w * v.w;
    }
    *(float4*)(red + rg * 2 * D + c0)     = s;
    *(float4*)(red + rg * 2 * D + D + c0) = s2;
    __syncthreads();
    // 128 threads: thread c reduces channel c over the 4 row-groups
    const int c = threadIdx.x;
    float ts = 0.f, ts2 = 0.f;
#pragma unroll
    for (int g = 0; g < 4; ++g) {
        ts  += red[g * 2 * D + c];
        ts2 += red[g * 2 * D + D + c];
    }
    atomic_add_f32(&sums[c], ts);
    atomic_add_f32(&sums[D + c], ts2);
}

// ---------------------------------------------------------------------------
// K6: BN affine + inverted dropout, in place on z (= d_out), float4/thread
// ---------------------------------------------------------------------------
__global__ void __launch_bounds__(256)
k_finalize(float* __restrict__ z, const float* __restrict__ sums,
           const float* __restrict__ gamma, const float* __restrict__ beta,
           const float* __restrict__ du) {
    const int i4 = blockIdx.x * blockDim.x + threadIdx.x;
    if (i4 >= (N_NODES * D) / 4) return;
    const int c0 = (i4 * 4) & (D - 1);
    const float inv_n = 1.0f / (float)N_NODES;
    const float4 zv = *(const float4*)(z + (size_t)i4 * 4);
    const float4 uv = *(const float4*)(du + (size_t)i4 * 4);
    const float4 sm = *(const float4*)(sums + c0);
    const float4 sq = *(const float4*)(sums + D + c0);
    const float4 gm = *(const float4*)(gamma + c0);
    const float4 bt = *(const float4*)(beta + c0);
    float4 o;
    {
        float mean = sm.x * inv_n, var = fmaf(sq.x, inv_n, -mean * mean);
        o.x = (zv.x - mean) * rsqrtf(var + BN_EPS) * gm.x + bt.x;
        o.x *= (uv.x > P_DROP) ? (1.0f / (1.0f - P_DROP)) : 0.0f;
    }
    {
        float mean = sm.y * inv_n, var = fmaf(sq.y, inv_n, -mean * mean);
        o.y = (zv.y - mean) * rsqrtf(var + BN_EPS) * gm.y + bt.y;
        o.y *= (uv.y > P_DROP) ? (1.0f / (1.0f - P_DROP)) : 0.0f;
    }
    {
        float mean = sm.z * inv_n, var = fmaf(sq.z, inv_n, -mean * mean);
        o.z = (zv.z - mean) * rsqrtf(var + BN_EPS) * gm.z + bt.z;
        o.z *= (uv.z > P_DROP) ? (1.0f / (1.0f - P_DROP)) : 0.0f;
    }
    {
        float mean = sm.w * inv_n, var = fmaf(sq.w, inv_n, -mean * mean);
        o.w = (zv.w - mean) * rsqrtf(var + BN_EPS) * gm.w + bt.w;
        o.w *= (uv.w > P_DROP) ? (1.0f / (1.0f - P_DROP)) : 0.0f;
    }
    *(float4*)(z + (size_t)i4 * 4) = o;
}

// ---------------------------------------------------------------------------
extern "C" void kernel_launch(void* const* d_in, const int* in_sizes, int n_in,
                              void* d_out, int out_size, void* d_ws, size_t ws_size,
                              hipStream_t stream) {
    const float*     x     = (const float*)d_in[0];
    const float*     w     = (const float*)d_in[1];
    const float*     bias  = (const float*)d_in[2];
    const float*     gamma = (const float*)d_in[3];
    const float*     beta  = (const float*)d_in[4];
    const float*     du    = (const float*)d_in[5];
    const long long* ei    = (const long long*)d_in[6];
    float* out = (float*)d_out;

    // workspace layout (floats): deg/dinv | sums(2*128) | y | acc  (~51.4 MB)
    float* deg  = (float*)d_ws;
    float* sums = deg + 50048;
    float* y    = sums + 256;
    float* acc  = y + (size_t)N_NODES * D;

    k_init<<<(N_NODES + 255) / 256, 256, 0, stream>>>(deg, sums);
    k_degree<<<(N_EDGES + 255) / 256, 256, 0, stream>>>(ei, deg);
    k_dinv<<<(N_NODES + 255) / 256, 256, 0, stream>>>(deg);

    const int m_tiles = N_NODES / 16;           // 3125
    k_gemm<<<(m_tiles + 7) / 8, 256, 0, stream>>>(x, w, deg, y, acc);

    k_edge_agg<<<(N_EDGES + 7) / 8, 256, 0, stream>>>(ei, y, acc);
    k_bn_stats<<<(N_NODES + BN_ROWS - 1) / BN_ROWS, 128, 0, stream>>>(acc, deg, bias, out, sums);
    k_finalize<<<((N_NODES * D) / 4 + 255) / 256, 256, 0, stream>>>(out, sums, gamma, beta, du);
}